// StillingerWeberLayer_8349416423611
// MI455X (gfx1250) — compile-verified
//
#include <hip/hip_runtime.h>
#include <hip/hip_bf16.h>

typedef __attribute__((ext_vector_type(2))) float v2f;
typedef __attribute__((ext_vector_type(8))) float v8f;

#define WAVES_PER_BLOCK 8
#define NBR 32

__global__ __launch_bounds__(256) void sw_energy_kernel(
    const float* __restrict__ coords,
    const float* __restrict__ Ap, const float* __restrict__ Bp,
    const float* __restrict__ pp, const float* __restrict__ qp,
    const float* __restrict__ sigp, const float* __restrict__ gamp,
    const float* __restrict__ cutp,
    const float* __restrict__ lamp, const float* __restrict__ cb0p,
    const float* __restrict__ cjkp,
    const int* __restrict__ elements, const int* __restrict__ nlist,
    float* __restrict__ out, int n_atoms)
{
    const int lane = threadIdx.x & 31;
    const int w    = threadIdx.x >> 5;
    const int atom = blockIdx.x * WAVES_PER_BLOCK + w;

    // Per-neighbor packed record: {r^2, 1/r, f=exp(g/(r-c)), bits(e | valid<<1)}
    __shared__ float4 shp[WAVES_PER_BLOCK][NBR];
    // Displacement components, planes x/y/z, for WMMA operand construction
    __shared__ float  shd[WAVES_PER_BLOCK][3][NBR];
    __shared__ float  bsum;

    float local = 0.0f;
    int eiv = 0;

    if (atom < n_atoms) {
        const int row = atom * (NBR + 1);
        const int ci  = nlist[row];
        const int nj  = nlist[row + 1 + lane];
        const float xi = coords[3*ci+0], yi = coords[3*ci+1], zi = coords[3*ci+2];
        const float dx = coords[3*nj+0] - xi;
        const float dy = coords[3*nj+1] - yi;
        const float dz = coords[3*nj+2] - zi;
        const float r2  = dx*dx + dy*dy + dz*dz;
        const float rij = sqrtf(r2);

        eiv = elements[row];
        const int ejv = elements[row + 1 + lane];
        const int s2  = eiv + ejv;

        const float A2 = Ap[s2], B2 = Bp[s2], P2 = pp[s2], Q2 = qp[s2];
        const float sg = sigp[s2], c2 = cutp[s2];
        const bool  m2 = rij < c2;

        // pair term: 0.5 * A*(B*sr^p - sr^q) * exp(sg/(r-c))
        {
            const float rr = m2 ? rij : 0.5f * c2;
            const float sr = sg / rr;
            const float E2 = A2 * (B2 * __powf(sr, P2) - __powf(sr, Q2)) *
                             __expf(sg / (rr - c2));
            local += m2 ? 0.5f * E2 : 0.0f;
        }
        // per-neighbor triplet factor: f_j = exp(gamma_ij/(r_ij - c_ij))
        const float gij  = gamp[s2];
        const float fj   = m2 ? __expf(gij / (rij - c2)) : 0.0f;
        const float rinv = 1.0f / rij;   // one divide per neighbor, amortized over ~31 triplets

        shd[w][0][lane] = dx;
        shd[w][1][lane] = dy;
        shd[w][2][lane] = dz;
        shp[w][lane] = make_float4(r2, rinv, fj,
                                   __int_as_float(ejv | (m2 ? 2 : 0)));
    }
    if (threadIdx.x == 0) bsum = 0.0f;
    __syncthreads();

    if (atom < n_atoms) {
        const int half = lane >> 4;   // K-split half of the WMMA operand
        const int nn   = lane & 15;   // row/col within a 16-group

        // 32-bit 16x4 A-operand layout (ISA 7.12.2): lanes 0-15 hold {K0=x,K1=y},
        // lanes 16-31 hold {K2=z,K3=0} of row M = lane%16. Gram symmetry makes
        // the B-operand (4x16) layout lane-identical. Branchless build:
        const float* db  = &shd[w][0][0];
        const int    xo  = half ? 2*NBR : 0;     // x-plane or z-plane
        const float  ysel = half ? 0.0f : 1.0f;  // K3 pad = 0 on hi half
        v2f G0, G1;
        G0[0] = db[xo + nn];
        G0[1] = db[NBR + nn] * ysel;
        G1[0] = db[xo + 16 + nn];
        G1[1] = db[NBR + 16 + nn] * ysel;

        const float lam0 = lamp[0], lam1 = lamp[1];
        const float cb00 = cb0p[0], cb01 = cb0p[1];
        const float cjk0 = cjkp[0], cjk1 = cjkp[1];

        const v8f zero = {};
        #pragma unroll
        for (int t = 0; t < 3; ++t) {
            const int rG = (t == 2) ? 1 : 0;   // row 16-group of Gram tile
            const int cG = (t == 0) ? 0 : 1;   // col 16-group (tile (1,0) is j>k)
            // Gram tile = D_r (16x4) * D_c^T (4x16), fp32 WMMA
            const v8f acc = __builtin_amdgcn_wmma_f32_16x16x4_f32(
                false, (rG ? G1 : G0), false, (cG ? G1 : G0),
                (short)0, zero, false, false);

            const int    kg  = cG * 16 + nn;   // global k (D-layout: N = lane%16)
            const float4 pk  = shp[w][kg];     // single ds_load_b128
            const float  r2k = pk.x, rinvk = pk.y, fk = pk.z;
            const int    epk = __float_as_int(pk.w);
            const int    ek  = epk & 1;
            const bool   vk  = (epk & 2) != 0;

            #pragma unroll
            for (int v = 0; v < 8; ++v) {
                const int jg = rG * 16 + half * 8 + v;   // D-layout: M=v (+8 hi lanes)
                const float4 pj = shp[w][jg];            // LDS broadcast within half-wave
                const int  epj = __float_as_int(pj.w);
                const int  ej  = epj & 1;
                bool ok = (jg < kg) && ((epj & 2) != 0) && vk &&
                          (eiv != ej) && (ej == ek);
                int ijk = 2 - (eiv + ej + ek);
                ijk = ijk < 0 ? 0 : (ijk > 1 ? 1 : ijk);
                const float cjkv = ijk ? cjk1 : cjk0;
                const float dot  = acc[v];
                // r_jk^2 = r_ij^2 + r_ik^2 - 2 d_j.d_k
                const float rjk2 = __builtin_fmaf(-2.0f, dot, pj.x + r2k);
                ok = ok && (rjk2 < cjkv * cjkv);
                const float cosb = dot * pj.y * rinvk;
                const float dd   = cosb - (ijk ? cb01 : cb00);
                const float e3   = (ijk ? lam1 : lam0) * pj.z * fk * dd * dd;
                local += ok ? e3 : 0.0f;                 // branchless select-accumulate
            }
        }
    }

    // wave32 reduction, then block, then one global atomic per block
    #pragma unroll
    for (int off = 16; off >= 1; off >>= 1)
        local += __shfl_xor(local, off, 32);
    if (lane == 0) atomicAdd(&bsum, local);
    __syncthreads();
    if (threadIdx.x == 0) atomicAdd(out, bsum);
}

extern "C" void kernel_launch(void* const* d_in, const int* in_sizes, int n_in,
                              void* d_out, int out_size, void* d_ws, size_t ws_size,
                              hipStream_t stream) {
    const float* coords   = (const float*)d_in[0];
    const float* A        = (const float*)d_in[1];
    const float* B        = (const float*)d_in[2];
    const float* p        = (const float*)d_in[3];
    const float* q        = (const float*)d_in[4];
    const float* sigma    = (const float*)d_in[5];
    const float* gamma_   = (const float*)d_in[6];
    const float* cutoff   = (const float*)d_in[7];
    const float* lam      = (const float*)d_in[8];
    const float* cosb0    = (const float*)d_in[9];
    const float* cutoffjk = (const float*)d_in[10];
    const int*   elements = (const int*)d_in[11];
    const int*   nl       = (const int*)d_in[12];

    const int n_atoms = in_sizes[12] / (NBR + 1);

    hipMemsetAsync(d_out, 0, (size_t)out_size * sizeof(float), stream);

    dim3 block(256);
    dim3 grid((n_atoms + WAVES_PER_BLOCK - 1) / WAVES_PER_BLOCK);
    hipLaunchKernelGGL(sw_energy_kernel, grid, block, 0, stream,
                       coords, A, B, p, q, sigma, gamma_, cutoff,
                       lam, cosb0, cutoffjk, elements, nl,
                       (float*)d_out, n_atoms);
}